// MIND_7310034338219
// MI455X (gfx1250) — compile-verified
//
#include <hip/hip_runtime.h>

#define KC 8
#define DD 64
#define ML 200
#define LP 208          // 13 * 16
#define ITERS 3
#define NEGINF (-1.0e9f)

typedef __attribute__((ext_vector_type(2))) float v2f;
typedef __attribute__((ext_vector_type(8))) float v8f;

// ---- LDS layout (float offsets) ----
constexpr int ES  = 68;    // E row stride (pad vs 64 banks)
constexpr int SCS = 64;    // Sc row stride
constexpr int WTS = 16;    // W transposed: [l][k]
constexpr int CS  = 68;    // caps row stride
constexpr int XS  = 132;   // MLP input row stride
constexpr int HS  = 264;   // hidden row stride

constexpr int F_SC   = 0;                 // LP*SCS = 13312
constexpr int F_E    = F_SC + LP * SCS;   // LP*ES  = 14144 (aliased below after phase 2)
constexpr int F_WT   = F_E;               // LP*WTS = 3328
constexpr int F_CAPS = F_WT + LP * WTS;   // 16*CS  = 1088
constexpr int F_X    = F_CAPS + 16 * CS;  // 16*XS  = 2112
constexpr int F_H    = F_X + 16 * XS;     // 16*HS  = 4224  (ends < F_E + LP*ES)
constexpr int F_BRT  = F_E + LP * ES;     // KC*LP  = 1664
constexpr int F_S    = F_BRT + KC * LP;   // 64*64  = 4096
constexpr int F_USER = F_S + DD * DD;     // 64
constexpr int F_NORM = F_USER + 64;       // 16
constexpr int F_IH   = F_NORM + 16;       // 208 ints
constexpr int F_IC   = F_IH + LP;
constexpr int F_IB   = F_IC + LP;
constexpr int F_KV   = F_IB + LP;
constexpr int F_TOT  = F_KV + 4;
constexpr size_t LDS_BYTES = (size_t)F_TOT * 4;   // ~135.7 KB -> 2 WGs per 320KB WGP

__device__ __forceinline__ v8f wmma4(v2f a, v2f b, v8f c) {
  // V_WMMA_F32_16X16X4_F32: (neg_a, A, neg_b, B, c_mod, C, reuse_a, reuse_b)
  return __builtin_amdgcn_wmma_f32_16x16x4_f32(false, a, false, b, (short)0, c,
                                               false, false);
}

__global__ void __launch_bounds__(256)
mind_kernel(const int* __restrict__ g_hist, const int* __restrict__ g_uid,
            const int* __restrict__ g_gen, const int* __restrict__ g_age,
            const float* __restrict__ g_Binit, const int* __restrict__ g_feats,
            const float* __restrict__ g_idTab, const float* __restrict__ g_catTab,
            const float* __restrict__ g_brTab, const float* __restrict__ g_uidTab,
            const float* __restrict__ g_genTab, const float* __restrict__ g_ageTab,
            const float* __restrict__ g_S, const float* __restrict__ g_W1,
            const float* __restrict__ g_b1, const float* __restrict__ g_W2,
            const float* __restrict__ g_b2,
            float* __restrict__ g_out, float* __restrict__ g_mask)
{
  extern __shared__ float sm[];
  float* sSc  = sm + F_SC;
  float* sE   = sm + F_E;
  float* sWt  = sm + F_WT;
  float* sCap = sm + F_CAPS;
  float* sX   = sm + F_X;
  float* sH   = sm + F_H;
  float* sBrt = sm + F_BRT;
  float* sS   = sm + F_S;
  float* sU   = sm + F_USER;
  float* sN   = sm + F_NORM;
  int*   sHist = (int*)(sm + F_IH);
  int*   sCat  = (int*)(sm + F_IC);
  int*   sBr   = (int*)(sm + F_IB);
  int*   sKv   = (int*)(sm + F_KV);

  const int b   = blockIdx.x;
  const int tid = threadIdx.x;
  const int wv  = tid >> 5;    // wave id (wave32)
  const int ln  = tid & 31;
  const int l16 = ln & 15;
  const int hx  = ln >> 4;     // lane half

  // ---- phase 0: stage ids / feature ids ----
  for (int l = tid; l < LP; l += 256) {
    int id = (l < ML) ? g_hist[b * ML + l] : 0;
    sHist[l] = id;
    sCat[l]  = g_feats[id * 2 + 0];
    sBr[l]   = g_feats[id * 2 + 1];
  }
  __syncthreads();

  if (tid == 0) {   // interest count -> Kv (consumed after next barrier)
    int cnt = 0;
    for (int l = 0; l < ML; ++l) cnt += (sHist[l] != 0) ? 1 : 0;
    float kvf = fminf(fmaxf(log2f((float)cnt), 1.0f), (float)KC);
    sKv[0] = (int)kvf;
  }

  // ---- phase 1: gather E, load S, user embeds, init Brt ----
  for (int idx = tid; idx < LP * DD; idx += 256) {
    int l = idx >> 6, c = idx & 63;
    float e = 0.0f;
    if (l < ML)
      e = (g_idTab[sHist[l] * DD + c] + g_catTab[sCat[l] * DD + c] +
           g_brTab[sBr[l] * DD + c]) * (1.0f / 3.0f);
    sE[l * ES + c] = e;
  }
  for (int idx = tid; idx < DD * DD; idx += 256) sS[idx] = g_S[idx];
  if (tid < 64) {
    float u;
    if (tid < 32)      u = g_uidTab[(long long)g_uid[b] * 32 + tid];
    else if (tid < 48) u = g_genTab[g_gen[b] * 16 + (tid - 32)];
    else               u = g_ageTab[g_age[b] * 16 + (tid - 48)];
    sU[tid] = u;
  }
  for (int idx = tid; idx < KC * LP; idx += 256) {
    int k = idx / LP, l = idx % LP;
    sBrt[idx] = (l < ML) ? g_Binit[k * ML + l] : 0.0f;
  }
  __syncthreads();

  // ---- phase 2: Sc = E @ S  ([LP x 64] = [LP x 64][64 x 64]) ----
  for (int t = wv; t < 13 * 4; t += 8) {
    int m0 = (t >> 2) * 16, n0 = (t & 3) * 16;
    v8f acc = {0, 0, 0, 0, 0, 0, 0, 0};
    for (int k0 = 0; k0 < DD; k0 += 4) {
      int ka = k0 + 2 * hx;
      v2f a, bf;
      a.x  = sE[(m0 + l16) * ES + ka + 0];
      a.y  = sE[(m0 + l16) * ES + ka + 1];
      bf.x = sS[(ka + 0) * DD + n0 + l16];
      bf.y = sS[(ka + 1) * DD + n0 + l16];
      acc = wmma4(a, bf, acc);
    }
    for (int v = 0; v < 8; ++v)
      sSc[(m0 + v + 8 * hx) * SCS + n0 + l16] = acc[v];
  }
  __syncthreads();

  // ---- phase 3: dynamic routing ----
  for (int it = 0; it < ITERS; ++it) {
    // 3a: softmax over capsule axis -> Wt[l][k] (rows 8..15 zero)
    for (int l = tid; l < LP; l += 256) {
      if (l < ML) {
        const int kv = sKv[0];
        const bool pad = (sHist[l] == 0);
        float x[KC]; float mx = NEGINF;
        for (int k = 0; k < KC; ++k) {
          float v = (pad || k >= kv) ? NEGINF : sBrt[k * LP + l];
          x[k] = v; mx = fmaxf(mx, v);
        }
        float s = 0.0f;
        for (int k = 0; k < KC; ++k) { float e = __expf(x[k] - mx); x[k] = e; s += e; }
        float inv = 1.0f / s;
        for (int k = 0; k < KC; ++k) sWt[l * WTS + k] = x[k] * inv;
        for (int k = KC; k < 16; ++k) sWt[l * WTS + k] = 0.0f;
      } else {
        for (int k = 0; k < 16; ++k) sWt[l * WTS + k] = 0.0f;
      }
    }
    __syncthreads();

    // 3b: caps_raw = W @ Sc  ([16 x 64] = [16 x LP][LP x 64])
    if (wv < 4) {
      int n0 = wv * 16;
      v8f acc = {0, 0, 0, 0, 0, 0, 0, 0};
      for (int k0 = 0; k0 < LP; k0 += 4) {
        int ka = k0 + 2 * hx;
        v2f a, bf;
        a.x  = sWt[(ka + 0) * WTS + l16];   // A[m][k] = Wt[k][m]
        a.y  = sWt[(ka + 1) * WTS + l16];
        bf.x = sSc[(ka + 0) * SCS + n0 + l16];
        bf.y = sSc[(ka + 1) * SCS + n0 + l16];
        acc = wmma4(a, bf, acc);
      }
      for (int v = 0; v < 8; ++v)
        sCap[(v + 8 * hx) * CS + n0 + l16] = acc[v];
    }
    __syncthreads();

    // 3c: squash
    if (tid < 16) {
      float sq = 0.0f;
      for (int c = 0; c < DD; ++c) { float v = sCap[tid * CS + c]; sq += v * v; }
      sN[tid] = sq / ((1.0f + sq) * sqrtf(sq) + 1e-9f);
    }
    __syncthreads();
    for (int idx = tid; idx < 16 * DD; idx += 256) {
      int r = idx >> 6, c = idx & 63;
      sCap[r * CS + c] *= sN[r];
    }
    __syncthreads();

    // 3d: Brt += caps @ Sc^T  (last-iteration update is unused in reference)
    if (it + 1 < ITERS) {
      for (int t = wv; t < 13; t += 8) {
        int n0 = t * 16;
        v8f acc = {0, 0, 0, 0, 0, 0, 0, 0};
        for (int k0 = 0; k0 < DD; k0 += 4) {
          int ka = k0 + 2 * hx;
          v2f a, bf;
          a.x  = sCap[l16 * CS + ka + 0];
          a.y  = sCap[l16 * CS + ka + 1];
          bf.x = sSc[(n0 + l16) * SCS + ka + 0];   // B[k][n] = Sc[n][k]
          bf.y = sSc[(n0 + l16) * SCS + ka + 1];
          acc = wmma4(a, bf, acc);
        }
        if (hx == 0) {  // rows m = v in 0..7 only
          for (int v = 0; v < 8; ++v)
            sBrt[v * LP + n0 + l16] += acc[v];
        }
      }
    }
    __syncthreads();
  }

  // ---- phase 4: MLP ----
  for (int idx = tid; idx < 16 * 128; idx += 256) {
    int r = idx >> 7, c = idx & 127;
    float v = 0.0f;
    if (r < KC) v = (c < DD) ? sCap[r * CS + c] : sU[c - DD];
    sX[r * XS + c] = v;
  }
  __syncthreads();

  // H = relu(X @ W1 + b1): [16 x 256]
  for (int t = wv; t < 16; t += 8) {
    int n0 = t * 16;
    v8f acc = {0, 0, 0, 0, 0, 0, 0, 0};
    for (int k0 = 0; k0 < 128; k0 += 4) {
      int ka = k0 + 2 * hx;
      v2f a, bf;
      a.x  = sX[l16 * XS + ka + 0];
      a.y  = sX[l16 * XS + ka + 1];
      bf.x = g_W1[(ka + 0) * 256 + n0 + l16];
      bf.y = g_W1[(ka + 1) * 256 + n0 + l16];
      acc = wmma4(a, bf, acc);
    }
    float bb = g_b1[n0 + l16];
    for (int v = 0; v < 8; ++v)
      sH[(v + 8 * hx) * HS + n0 + l16] = fmaxf(acc[v] + bb, 0.0f);
  }
  __syncthreads();

  // Out = H @ W2 + b2: [16 x 64], keep capsule rows 0..7
  if (wv < 4) {
    int n0 = wv * 16;
    v8f acc = {0, 0, 0, 0, 0, 0, 0, 0};
    for (int k0 = 0; k0 < 256; k0 += 4) {
      int ka = k0 + 2 * hx;
      v2f a, bf;
      a.x  = sH[l16 * HS + ka + 0];
      a.y  = sH[l16 * HS + ka + 1];
      bf.x = g_W2[(ka + 0) * DD + n0 + l16];
      bf.y = g_W2[(ka + 1) * DD + n0 + l16];
      acc = wmma4(a, bf, acc);
    }
    if (hx == 0) {
      float bb = g_b2[n0 + l16];
      for (int v = 0; v < 8; ++v)
        g_out[((size_t)b * KC + v) * DD + n0 + l16] = acc[v] + bb;
    }
  }
  if (tid < KC) g_mask[(size_t)b * KC + tid] = (tid >= sKv[0]) ? 1.0f : 0.0f;
}

extern "C" void kernel_launch(void* const* d_in, const int* in_sizes, int n_in,
                              void* d_out, int out_size, void* d_ws, size_t ws_size,
                              hipStream_t stream) {
  const int*   hist   = (const int*)  d_in[0];
  const int*   uid    = (const int*)  d_in[1];
  const int*   gen    = (const int*)  d_in[2];
  const int*   age    = (const int*)  d_in[3];
  const float* Binit  = (const float*)d_in[4];
  const int*   feats  = (const int*)  d_in[5];
  const float* idTab  = (const float*)d_in[6];
  const float* catTab = (const float*)d_in[7];
  const float* brTab  = (const float*)d_in[8];
  const float* uidTab = (const float*)d_in[9];
  const float* genTab = (const float*)d_in[10];
  const float* ageTab = (const float*)d_in[11];
  const float* Smat   = (const float*)d_in[12];
  const float* W1     = (const float*)d_in[13];
  const float* b1     = (const float*)d_in[14];
  const float* W2     = (const float*)d_in[15];
  const float* b2     = (const float*)d_in[16];

  const int B = in_sizes[1];                 // 4096 (user_id count)
  float* outp    = (float*)d_out;            // [B, 8, 64] f32
  float* maskOut = outp + (size_t)B * KC * DD;  // cap_mask [B, 8] as 0/1 floats

  (void)hipFuncSetAttribute(reinterpret_cast<const void*>(mind_kernel),
                            hipFuncAttributeMaxDynamicSharedMemorySize,
                            (int)LDS_BYTES);

  mind_kernel<<<B, 256, LDS_BYTES, stream>>>(
      hist, uid, gen, age, Binit, feats, idTab, catTab, brTab,
      uidTab, genTab, ageTab, Smat, W1, b1, W2, b2, outp, maskOut);
}